// ADJ_AttCenter_14740327759982
// MI455X (gfx1250) — compile-verified
//
#include <hip/hip_runtime.h>
#include <hip/hip_bf16.h>
#include <math.h>

// ---------------------------------------------------------------------------
// ADJ_AttCenter for MI455X (gfx1250, wave32).
//   out = [ x_new (N+4)xC | adj (N+4)x(N+4) ]  (float32)
// adj is computed analytically: deg(regular)=2, deg(center)=s+4.
// Scores GEMV uses V_WMMA_F32_16X16X4_F32.
// ---------------------------------------------------------------------------

typedef __attribute__((ext_vector_type(2))) float v2f;
typedef __attribute__((ext_vector_type(4))) float v4f;
typedef __attribute__((ext_vector_type(8))) float v8f;

// ---- workspace zeroing (segsum[4] + centers[4*C]) --------------------------
__global__ void adjac_init_ws(float* __restrict__ ws, int nfloats) {
    int i = blockIdx.x * blockDim.x + threadIdx.x;
    if (i < nfloats) ws[i] = 0.0f;
}

// ---- scores: w = x @ att_w^T + b, per-wave 16 rows via WMMA f32 16x16x4 ----
// A tile (16x4 f32): lanes 0-15 -> M=lane, K=k+{0,1}; lanes 16-31 -> K=k+{2,3}
// B tile (4x16 f32): every N column = att_w[k..k+3] (same K striping as A),
// so D[m,n] = score(row m) for all n; lanes 0 and 16 read column N=0.
__global__ __launch_bounds__(128) void adjac_scores_wmma(
    const float* __restrict__ x, const float* __restrict__ attw,
    const float* __restrict__ attb, float* __restrict__ w,
    float* __restrict__ segsum, int C_, int N_, int S_) {
    const int wave    = threadIdx.x >> 5;
    const int lane    = threadIdx.x & 31;
    const int rowbase = (blockIdx.x * 4 + wave) * 16;
    if (rowbase >= N_) return;              // uniform per wave: EXEC stays full

    const int m    = lane & 15;
    const int koff = (lane >> 4) * 2;       // lane half selects K offset 0 / 2
    const float* arow = x + (size_t)(rowbase + m) * C_ + koff;

    v8f acc = {};
    #pragma unroll 4
    for (int k = 0; k < C_; k += 4) {
        v2f a = *(const v2f*)(arow + k);            // A: x[row, k+koff .. +1]
        v2f b = *(const v2f*)(attw + k + koff);     // B: att_w broadcast over N
        acc = __builtin_amdgcn_wmma_f32_16x16x4_f32(
            /*neg_a=*/false, a, /*neg_b=*/false, b,
            /*c_mod=*/(short)0, acc, /*reuse_a=*/false, /*reuse_b=*/false);
    }

    // D column N=0 lives in lane 0 (M=0..7) and lane 16 (M=8..15)
    if ((lane & 15) == 0) {
        const int   r0   = rowbase + (lane >> 4) * 8;
        const float bias = attb[0];
        float part = 0.0f;
        #pragma unroll
        for (int v = 0; v < 8; ++v) {
            float sc = acc[v] + bias;
            w[r0 + v] = sc;
            part += sc;
        }
        atomicAdd(&segsum[rowbase / S_], part);
    }
}

// ---- centers accumulation: centers[seg] += sum_r w[r] * x[r, :] ------------
__global__ __launch_bounds__(256) void adjac_centers_accum(
    const float* __restrict__ x, const float* __restrict__ w,
    float* __restrict__ centers, int C_, int S_, int rowsPerBlock) {
    const int seg = blockIdx.z;
    const int col = blockIdx.x * 256 + threadIdx.x;
    const int r0  = seg * S_ + blockIdx.y * rowsPerBlock;
    float acc = 0.0f;
    for (int r = 0; r < rowsPerBlock; ++r) {
        const int row = r0 + r;
        acc += w[row] * x[(size_t)row * C_ + col];
    }
    atomicAdd(&centers[seg * C_ + col], acc);
}

// ---- finalize: out rows N..N+3 = centers/segsum ----------------------------
__global__ void adjac_finalize_centers(
    const float* __restrict__ centers, const float* __restrict__ segsum,
    float* __restrict__ out, int C_, int N_) {
    int i = blockIdx.x * blockDim.x + threadIdx.x;
    if (i < 4 * C_) {
        int seg = i / C_;
        out[(size_t)(N_ + seg) * C_ + (i - seg * C_)] = centers[i] / segsum[seg];
    }
}

// ---- copy x -> out[0 : N*C]  (NT stores: don't evict x from L2) ------------
__global__ __launch_bounds__(256) void adjac_copy_x(
    const v4f* __restrict__ x4, v4f* __restrict__ out4, int n4) {
    int i = blockIdx.x * 256 + threadIdx.x;
    if (i < n4) {
        v4f v = x4[i];
        __builtin_nontemporal_store(v, &out4[i]);
    }
}

// ---- adj: D * A * D written analytically, v4f NT streaming stores ----------
// row < N : adj[row,row]=0.5 ; adj[row, N+row/s]=c1
// row >= N: adj[row, seg*s .. ) = c1 over its segment; adj[row, N..N+3] = c2
__global__ __launch_bounds__(256) void adjac_build_adj(
    float* __restrict__ adj, unsigned total4, unsigned row4 /*=(N+4)/4*/,
    unsigned N_, int sShift, float c1, float c2) {
    unsigned idx = blockIdx.x * 256u + threadIdx.x;
    if (idx >= total4) return;
    unsigned row  = idx / row4;
    unsigned col0 = (idx - row * row4) * 4u;

    v4f o;
    #pragma unroll
    for (int c = 0; c < 4; ++c) {
        unsigned col = col0 + (unsigned)c;
        float val = 0.0f;
        if (row < N_) {
            if (col == row)                                      val = 0.5f;
            else if (col >= N_ && (col - N_) == (row >> sShift)) val = c1;
        } else {
            if (col >= N_)                                       val = c2;
            else if ((col >> sShift) == (row - N_))              val = c1;
        }
        o[c] = val;
    }
    __builtin_nontemporal_store(o, reinterpret_cast<v4f*>(adj) + idx);
}

// ---------------------------------------------------------------------------
extern "C" void kernel_launch(void* const* d_in, const int* in_sizes, int n_in,
                              void* d_out, int out_size, void* d_ws, size_t ws_size,
                              hipStream_t stream) {
    const float* x    = (const float*)d_in[0];
    const float* attw = (const float*)d_in[1];
    const float* attb = (const float*)d_in[2];

    const int C_ = in_sizes[1];            // 2048
    const int N_ = in_sizes[0] / C_;       // 8192
    const int S_ = N_ / 4;                 // 2048
    const int M_ = N_ + 4;                 // 8196

    int sShift = 0;                        // log2(S_) ; S_ is a power of two
    while ((1 << sShift) < S_) ++sShift;

    const float c2 = (float)(1.0 / (double)(S_ + 4));                 // 1/2052
    const float c1 = (float)(1.0 / sqrt(2.0 * (double)(S_ + 4)));     // 1/sqrt(4104)

    // workspace layout (floats): segsum[4] | centers[4*C] | w[N]
    float* ws      = (float*)d_ws;
    float* segsum  = ws;
    float* centers = ws + 4;
    float* wbuf    = ws + 4 + 4 * C_;
    const int nzero = 4 + 4 * C_;

    float* out_x   = (float*)d_out;
    float* out_adj = (float*)d_out + (size_t)M_ * C_;

    // 1) zero accumulators (must happen every call; graph replays this)
    adjac_init_ws<<<(nzero + 255) / 256, 256, 0, stream>>>(ws, nzero);

    // 2) scores via WMMA f32 16x16x4 (4 waves/block, 16 rows/wave)
    {
        int tiles  = N_ / 16;              // 512
        int blocks = (tiles + 3) / 4;      // 128
        adjac_scores_wmma<<<blocks, 128, 0, stream>>>(x, attw, attb, wbuf, segsum, C_, N_, S_);
    }

    // 3) unnormalized weighted row-sums per segment
    {
        const int rowsPerBlock = 128;
        dim3 grid(C_ / 256, S_ / rowsPerBlock, 4);   // (8, 16, 4)
        adjac_centers_accum<<<grid, 256, 0, stream>>>(x, wbuf, centers, C_, S_, rowsPerBlock);
    }

    // 4) normalize centers into out rows N..N+3
    adjac_finalize_centers<<<(4 * C_ + 255) / 256, 256, 0, stream>>>(centers, segsum, out_x, C_, N_);

    // 5) stream-copy x into out rows 0..N-1
    {
        int n4 = (N_ * C_) / 4;
        adjac_copy_x<<<(n4 + 255) / 256, 256, 0, stream>>>(
            (const v4f*)x, (v4f*)out_x, n4);
    }

    // 6) analytic adjacency, v4f NT streaming stores
    {
        unsigned row4   = (unsigned)(M_ / 4);            // 2049
        unsigned total4 = (unsigned)M_ * row4;           // 16,793,604
        unsigned blocks = (total4 + 255u) / 256u;        // 65,601
        adjac_build_adj<<<blocks, 256, 0, stream>>>(out_adj, total4, row4,
                                                    (unsigned)N_, sShift, c1, c2);
    }
}